// QuantumKernelLayer_65481071407684
// MI455X (gfx1250) — compile-verified
//
#include <hip/hip_runtime.h>
#include <hip/hip_bf16.h>
#include <math.h>

// Quantum fidelity kernel  K[b,c] = | prod_j cos((x_bj - c_cj)/2) |
// Rewritten as GEMM:  K = | F @ G^T |,  F:[B,256], G:[C,256] f32 feature maps
//   F[b,k] = prod_j ( bit_j(k) ? sin(x_bj/2) : cos(x_bj/2) )
// GEMM runs on V_WMMA_F32_16X16X4_F32 (exact f32 matrix path on CDNA5).

typedef __attribute__((ext_vector_type(2))) float v2f;
typedef __attribute__((ext_vector_type(8))) float v8f;

#define NWIRES 8
#define KDIM   256   // 2^NWIRES

// One block per input row; thread k (0..255) builds feature F[row][k].
__global__ __launch_bounds__(KDIM) void qk_features(const float* __restrict__ v,
                                                    float* __restrict__ feat) {
    const int row = blockIdx.x;
    const int k   = threadIdx.x;            // subset index, bits select sin vs cos
    const float* p = v + (size_t)row * NWIRES;
    float prod = 1.0f;
#pragma unroll
    for (int j = 0; j < NWIRES; ++j) {
        float a = 0.5f * p[j];              // broadcast read, L2/L0 resident
        float cj = __cosf(a) * 0.0f + cosf(a);   // keep precise libm cos
        float sj = sinf(a);
        prod *= ((k >> j) & 1) ? sj : cj;
    }
    feat[(size_t)row * KDIM + k] = prod;    // coalesced row-major store
}

// GEMM: out[b,c] = | sum_k F[b,k] * G[c,k] |
// Block = 256 threads = 8 waves; each wave computes a 32x32 output tile
// as 2x2 WMMA 16x16 tiles, K-loop in steps of 4 (V_WMMA_F32_16X16X4_F32).
//
// f32 A-fragment layout (ISA 7.12.2, 16x4): lanes 0-15 carry M=lane with
// {VGPR0,VGPR1} = {K+0,K+1}; lanes 16-31 carry {K+2,K+3}.  -> one float2
// load per lane at column offset (k + 2*(lane>>4)).  B (4x16) mirrors this
// with N = lane&15, since G is accessed as G[c0+n][k] (B[k][n] = G^T).
__global__ __launch_bounds__(256) void qk_wmma_gemm(const float* __restrict__ F,
                                                    const float* __restrict__ G,
                                                    float* __restrict__ out,
                                                    int Cn) {
    const int lane = threadIdx.x & 31;
    const int wave = threadIdx.x >> 5;
    const int l16  = lane & 15;
    const int half = lane >> 4;                 // 0: K0/K1 lanes, 1: K2/K3 lanes

    const int c0 = blockIdx.x * 32;             // N tile base (centers)
    const int b0 = (blockIdx.y * 8 + wave) * 32;// M tile base (batch)

    const float* A0 = F + (size_t)(b0 + l16)      * KDIM + 2 * half;
    const float* A1 = F + (size_t)(b0 + 16 + l16) * KDIM + 2 * half;
    const float* B0 = G + (size_t)(c0 + l16)      * KDIM + 2 * half;
    const float* B1 = G + (size_t)(c0 + 16 + l16) * KDIM + 2 * half;

    v8f acc00 = {}, acc01 = {}, acc10 = {}, acc11 = {};

#pragma unroll 8
    for (int k = 0; k < KDIM; k += 4) {
        v2f a0 = *(const v2f*)(A0 + k);
        v2f a1 = *(const v2f*)(A1 + k);
        v2f g0 = *(const v2f*)(B0 + k);
        v2f g1 = *(const v2f*)(B1 + k);
        acc00 = __builtin_amdgcn_wmma_f32_16x16x4_f32(false, a0, false, g0,
                                                      (short)0, acc00, false, false);
        acc01 = __builtin_amdgcn_wmma_f32_16x16x4_f32(false, a0, false, g1,
                                                      (short)0, acc01, false, false);
        acc10 = __builtin_amdgcn_wmma_f32_16x16x4_f32(false, a1, false, g0,
                                                      (short)0, acc10, false, false);
        acc11 = __builtin_amdgcn_wmma_f32_16x16x4_f32(false, a1, false, g1,
                                                      (short)0, acc11, false, false);
    }

    // C/D layout: VGPR v holds row m = v + 8*half, col n = lane&15.
    const int n0 = c0 + l16;
    const int n1 = c0 + 16 + l16;
#pragma unroll
    for (int v = 0; v < 8; ++v) {
        const int m = v + 8 * half;
        out[(size_t)(b0 + m)      * Cn + n0] = fabsf(acc00[v]);
        out[(size_t)(b0 + m)      * Cn + n1] = fabsf(acc01[v]);
        out[(size_t)(b0 + 16 + m) * Cn + n0] = fabsf(acc10[v]);
        out[(size_t)(b0 + 16 + m) * Cn + n1] = fabsf(acc11[v]);
    }
}

extern "C" void kernel_launch(void* const* d_in, const int* in_sizes, int n_in,
                              void* d_out, int out_size, void* d_ws, size_t ws_size,
                              hipStream_t stream) {
    const float* x       = (const float*)d_in[0];   // [B, 8] f32
    const float* centers = (const float*)d_in[1];   // [C, 8] f32
    float* out = (float*)d_out;                     // [B, C] f32

    const int B = in_sizes[0] / NWIRES;             // 8192
    const int C = in_sizes[1] / NWIRES;             // 2048

    // Workspace: F [B,256] then G [C,256], f32. 8 MB + 2 MB — L2 resident.
    float* Ffeat = (float*)d_ws;
    float* Gfeat = Ffeat + (size_t)B * KDIM;

    qk_features<<<dim3(B), dim3(KDIM), 0, stream>>>(x, Ffeat);
    qk_features<<<dim3(C), dim3(KDIM), 0, stream>>>(centers, Gfeat);

    // Wave tile 32x32, 8 waves/block -> block covers 256 rows x 32 cols.
    dim3 grid(C / 32, B / (32 * 8));
    qk_wmma_gemm<<<grid, dim3(256), 0, stream>>>(Ffeat, Gfeat, out, C);
}